// FrustumClusterProposer_29025388987076
// MI455X (gfx1250) — compile-verified
//
#include <hip/hip_runtime.h>
#include <hip/hip_bf16.h>
#include <math.h>

// ---------------------------------------------------------------------------
// FrustumClusterProposer on gfx1250 (MI455X)
//
// Phase A (massively parallel, WMMA): Gram matrix G = P*P^T via
//   V_WMMA_F32_16X16X4_F32 (K=3 padded to 4), converted in-register to the
//   pairwise dist/angle matrices (75.5 MB -> lives in the 192MB L2).
// Phase B (latency-bound): 3071-step greedy scan in ONE 1024-thread block,
//   all state in LDS (~58 KB of the 320KB WGP budget), wave32 shfl reductions.
// Phase C: segment stats, bitonic argsort, sequential merge, final remap.
// ---------------------------------------------------------------------------

#define NPTS 3072
#define NTH  1024
#define PER  (NPTS / NTH)   // 3 elements per thread in the scan block
#define ANGLE_THR 1.5707963267948966f

typedef __attribute__((ext_vector_type(2))) float v2f;
typedef __attribute__((ext_vector_type(8))) float v8f;

// ---------------- wave32 reductions ----------------
__device__ __forceinline__ float wredMin(float v){
  for (int o = 16; o > 0; o >>= 1) v = fminf(v, __shfl_down(v, o, 32));
  return v;
}
__device__ __forceinline__ float wredMax(float v){
  for (int o = 16; o > 0; o >>= 1) v = fmaxf(v, __shfl_down(v, o, 32));
  return v;
}
__device__ __forceinline__ float wredSum(float v){
  for (int o = 16; o > 0; o >>= 1) v += __shfl_down(v, o, 32);
  return v;
}
__device__ __forceinline__ unsigned long long wredMinU64(unsigned long long v){
  for (int o = 16; o > 0; o >>= 1){
    unsigned long long b = __shfl_down(v, o, 32);
    v = (b < v) ? b : v;
  }
  return v;
}

// ---------------- Phase 0: per-point norms ----------------
__global__ void prep_norms(const float* __restrict__ pts,
                           float* __restrict__ n2, float* __restrict__ nrm, int n){
  int i = blockIdx.x * blockDim.x + threadIdx.x;
  if (i < n){
    float x = pts[3*i], y = pts[3*i+1], z = pts[3*i+2];
    float s = x*x + y*y + z*z;
    n2[i] = s;
    nrm[i] = sqrtf(s);
  }
}

// ---------------- Phase A: pairwise dist/angle tiles via WMMA f32 16x16x4 ---
// One wave per 16x16 output tile. A = rows R..R+15 of P (16x4, K=x,y,z,0),
// B = rows Cb..Cb+15 of P transposed (4x16). D = Gram tile.
__global__ __launch_bounds__(256)
void pair_wmma(const float* __restrict__ pts,
               const float* __restrict__ n2, const float* __restrict__ nrm,
               float* __restrict__ dists, float* __restrict__ angles, int n){
  int lane = threadIdx.x & 31;
  int wave = threadIdx.x >> 5;
  int tc = blockIdx.x;                 // tile column
  int tr = blockIdx.y * 8 + wave;      // tile row
  int R  = tr * 16, Cb = tc * 16;
  int m  = lane & 15;
  int hi = lane >> 4;

  v8f c = {};
#if __has_builtin(__builtin_amdgcn_wmma_f32_16x16x4_f32)
  // A: 16x4, lanes 0-15 hold K={0,1}, lanes 16-31 hold K={2,3} (K=3 padded 0)
  const float* pa = pts + (size_t)(R + m) * 3;
  v2f a, b;
  a.x = hi ? pa[2] : pa[0];
  a.y = hi ? 0.0f  : pa[1];
  // B: 4x16, half-wave split mirrors C/D row split: lanes0-15 K={0,1}, 16-31 K={2,3}
  const float* pb = pts + (size_t)(Cb + m) * 3;
  b.x = hi ? pb[2] : pb[0];
  b.y = hi ? 0.0f  : pb[1];
  c = __builtin_amdgcn_wmma_f32_16x16x4_f32(false, a, false, b, (short)0, c,
                                            false, false);
#else
  // scalar fallback (should not trigger on gfx1250 toolchains)
  for (int v = 0; v < 8; ++v){
    int i = R + v + 8*hi, j = Cb + m;
    c[v] = pts[3*i]*pts[3*j] + pts[3*i+1]*pts[3*j+1] + pts[3*i+2]*pts[3*j+2];
  }
#endif

  int j = Cb + m;
  float n2j = n2[j];
  for (int v = 0; v < 8; ++v){
    int i = R + v + 8*hi;                 // C/D layout: VGPR v, half-wave -> M
    float g   = c[v];
    float n2i = n2[i];
    float d2  = n2i + n2j - 2.0f * g;
    float dist = sqrtf(fmaxf(d2, 0.0f));
    float dotv = n2i - g;                 // -p_i . (p_j - p_i)
    float den  = nrm[i] * dist + 1e-8f;
    float ang  = acosf(fminf(fmaxf(dotv / den, -1.0f), 1.0f));
    if (i == j){ dist = 0.0f; ang = acosf(0.0f); }  // match exact-zero diagonal
    dists [(size_t)i * n + j] = dist;
    angles[(size_t)i * n + j] = ang;
  }
}

// ---------------- Phase B: sequential greedy scan (one block) ----------------
template<bool PRE>
__global__ __launch_bounds__(1024)
void scan_kernel(const float* __restrict__ pts, const int* __restrict__ lbl,
                 const float* __restrict__ anchors,
                 int* __restrict__ dlOut, int* __restrict__ ccOut,
                 int* __restrict__ meta,
                 const float* __restrict__ dists, const float* __restrict__ angles){
  __shared__ float sPx[NPTS], sPy[NPTS], sPz[NPTS];
  __shared__ int   sDl[NPTS];
  __shared__ unsigned char sCls[NPTS];
  __shared__ short sCc[NPTS];
  __shared__ float rMin[32], rMax[32], rAmax[32], rSx[32], rSy[32], rSz[32], rCnt[32];
  __shared__ unsigned long long rKey[32];
  __shared__ float aLW[10], aH[10], aR2[10];
  __shared__ float bcf[8];
  __shared__ int   bci[4];

  int t = threadIdx.x, lane = t & 31, wv = t >> 5;
  const float INF = __builtin_inff();

  for (int j = t; j < NPTS; j += NTH){
    sPx[j] = pts[3*j]; sPy[j] = pts[3*j+1]; sPz[j] = pts[3*j+2];
    sDl[j] = -1; sCls[j] = (unsigned char)lbl[j]; sCc[j] = -1;
  }
  if (t < 10){
    float l = anchors[3*t], w = anchors[3*t+1], h = anchors[3*t+2];
    aLW[t] = fmaxf(l, w); aH[t] = h; aR2[t] = sqrtf(l*l + w*w + h*h) * 0.5f;
  }
  if (t == 0){ sDl[0] = 0; bci[0] = 0; bci[1] = 0; }   // cur=0, lab=0
  __syncthreads();

  for (int s = 0; s < NPTS - 1; ++s){
    int cur = bci[0], lab = bci[1];
    int clsCur = sCls[cur];
    float pcx = sPx[cur], pcy = sPy[cur], pcz = sPz[cur];
    if (t == 0 && sCc[lab] < 0) sCc[lab] = (short)clsCur;
    float nrmCur = PRE ? 0.0f : sqrtf(pcx*pcx + pcy*pcy + pcz*pcz);
    const float* drow = PRE ? (dists  + (size_t)cur * NPTS) : nullptr;
    const float* arow = PRE ? (angles + (size_t)cur * NPTS) : nullptr;

    // pass 1: dmin/dmax over unvisited, angle-max + centroid over current cluster
    float dmin = INF, dmax = -INF, amax = -INF, sx = 0, sy = 0, sz = 0, cnt = 0;
    for (int u = 0; u < PER; ++u){
      int j = t + u * NTH;
      int lj = sDl[j];
      float dist;
      if (PRE) dist = drow[j];
      else {
        float dx = sPx[j]-pcx, dy = sPy[j]-pcy, dz = sPz[j]-pcz;
        dist = sqrtf(dx*dx + dy*dy + dz*dz);
      }
      if (lj == -1){
        dmin = fminf(dmin, dist); dmax = fmaxf(dmax, dist);
      } else if (lj == lab){
        float ang;
        if (PRE) ang = arow[j];
        else {
          float dx = sPx[j]-pcx, dy = sPy[j]-pcy, dz = sPz[j]-pcz;
          float dotv = -(pcx*dx + pcy*dy + pcz*dz);
          float den  = nrmCur * dist + 1e-8f;
          ang = acosf(fminf(fmaxf(dotv / den, -1.0f), 1.0f));
        }
        amax = fmaxf(amax, ang);
        sx += sPx[j]; sy += sPy[j]; sz += sPz[j]; cnt += 1.0f;
      }
    }
    dmin = wredMin(dmin); dmax = wredMax(dmax); amax = wredMax(amax);
    sx = wredSum(sx); sy = wredSum(sy); sz = wredSum(sz); cnt = wredSum(cnt);
    if (lane == 0){
      rMin[wv]=dmin; rMax[wv]=dmax; rAmax[wv]=amax;
      rSx[wv]=sx; rSy[wv]=sy; rSz[wv]=sz; rCnt[wv]=cnt;
    }
    __syncthreads();
    if (wv == 0){
      float a = wredMin(rMin[lane]);
      float b = wredMax(rMax[lane]);
      float c = wredMax(rAmax[lane]);
      float x = wredSum(rSx[lane]);
      float y = wredSum(rSy[lane]);
      float z = wredSum(rSz[lane]);
      float k = wredSum(rCnt[lane]);
      if (lane == 0){
        bcf[0]=a; bcf[1]=b; bcf[2]=c;
        bcf[3]=x/k; bcf[4]=y/k; bcf[5]=z/k;
      }
    }
    __syncthreads();

    // pass 2: argmin over unvisited of dcost + lcost (first-index tiebreak)
    float Dmin = bcf[0];
    float inv  = 1.0f / (bcf[1] - Dmin + 1e-8f);
    unsigned long long best = ~0ull;
    for (int u = 0; u < PER; ++u){
      int j = t + u * NTH;
      if (sDl[j] == -1){
        float dist;
        if (PRE) dist = drow[j];
        else {
          float dx = sPx[j]-pcx, dy = sPy[j]-pcy, dz = sPz[j]-pcz;
          dist = sqrtf(dx*dx + dy*dy + dz*dz);
        }
        float dcost = (dist - Dmin) * inv;
        float lcost = fminf(fabsf((float)(int)sCls[j] - (float)clsCur), 1.0f);
        float cost = dcost + lcost;                 // >= 0, finite
        unsigned long long key =
            ((unsigned long long)__float_as_uint(cost) << 32) | (unsigned)j;
        best = (key < best) ? key : best;
      }
    }
    best = wredMinU64(best);
    if (lane == 0) rKey[wv] = best;
    __syncthreads();
    if (t == 0){
      unsigned long long b = rKey[0];
      for (int i = 1; i < 32; ++i) b = (rKey[i] < b) ? rKey[i] : b;
      int ci = (int)(b & 0xffffffffu);
      float dx = pcx - sPx[ci], dy = pcy - sPy[ci], dz = pcz - sPz[ci];
      float lw = aLW[clsCur], hh = aH[clsCur], r2 = aR2[clsCur];
      bool nc = (fabsf(dx) > lw) || (fabsf(dy) > lw) || (fabsf(dz) > hh);
      nc = nc || (bcf[2] < ANGLE_THR);
      nc = nc || ((int)sCls[ci] != clsCur);
      nc = nc || (sqrtf(dx*dx + dy*dy + dz*dz) > r2);
      float ex = bcf[3]-sPx[ci], ey = bcf[4]-sPy[ci], ez = bcf[5]-sPz[ci];
      nc = nc || (sqrtf(ex*ex + ey*ey + ez*ez) > r2);
      lab += nc ? 1 : 0;
      sDl[ci] = lab;
      bci[0] = ci; bci[1] = lab;
    }
    __syncthreads();
  }

  for (int j = t; j < NPTS; j += NTH){ dlOut[j] = sDl[j]; ccOut[j] = (int)sCc[j]; }
  if (t == 0) meta[0] = bci[1];   // num_clusters = max label = final lab
}

// ---------------- Phase C: stats / sort / merge / writeout ----------------
__global__ void zero_init(int* counts, float* sx, float* sy, float* sz,
                          int* remap, int n){
  int j = blockIdx.x * blockDim.x + threadIdx.x;
  if (j < n){ counts[j]=0; sx[j]=0.f; sy[j]=0.f; sz[j]=0.f; remap[j]=j; }
}

__global__ void stats_accum(const int* __restrict__ dl, const float* __restrict__ pts,
                            int* counts, float* sx, float* sy, float* sz, int n){
  int j = blockIdx.x * blockDim.x + threadIdx.x;
  if (j < n){
    int c = dl[j];
    atomicAdd(&counts[c], 1);
    atomicAdd(&sx[c], pts[3*j]);
    atomicAdd(&sy[c], pts[3*j+1]);
    atomicAdd(&sz[c], pts[3*j+2]);
  }
}

__global__ void centres_fin(const int* counts, const float* sx, const float* sy,
                            const float* sz, float* cx, float* cy, float* cz, int n){
  int c = blockIdx.x * blockDim.x + threadIdx.x;
  if (c < n){
    float k = fmaxf((float)counts[c], 1.0f);
    cx[c] = sx[c]/k; cy[c] = sy[c]/k; cz[c] = sz[c]/k;
  }
}

// Stable descending argsort of masked counts == ascending bitonic sort of
// composite keys (0x7fffffff - maskedCount, index).
__global__ __launch_bounds__(1024)
void sort_kernel(const int* __restrict__ counts, const int* __restrict__ meta,
                 int* __restrict__ order, int n){
  __shared__ unsigned long long key[4096];
  int numC = meta[0];
  for (int j = threadIdx.x; j < 4096; j += 1024){
    if (j < n){
      int mc = (j < numC) ? counts[j] : -1;
      unsigned int k32 = 0x7fffffffu - (unsigned int)mc;   // -1 -> 0x80000000 (last)
      key[j] = ((unsigned long long)k32 << 32) | (unsigned)j;
    } else key[j] = ~0ull;
  }
  __syncthreads();
  for (int k = 2; k <= 4096; k <<= 1){
    for (int j2 = k >> 1; j2 > 0; j2 >>= 1){
      for (int i = threadIdx.x; i < 4096; i += 1024){
        int ixj = i ^ j2;
        if (ixj > i){
          unsigned long long a = key[i], b = key[ixj];
          bool up = ((i & k) == 0);
          if ((up && a > b) || (!up && a < b)){ key[i] = b; key[ixj] = a; }
        }
      }
      __syncthreads();
    }
  }
  for (int j = threadIdx.x; j < n; j += 1024)
    order[j] = (int)(key[j] & 0xffffffffu);
}

__global__ __launch_bounds__(1024)
void merge_kernel(const int* __restrict__ order,
                  const float* __restrict__ cx, const float* __restrict__ cy,
                  const float* __restrict__ cz, const int* __restrict__ cc,
                  const float* __restrict__ anchors, const int* __restrict__ meta,
                  int* __restrict__ remap){
  __shared__ float oCx[NPTS], oCy[NPTS], oCz[NPTS];
  __shared__ short oCls[NPTS];
  __shared__ unsigned char keep[NPTS];
  __shared__ int sOrder[NPTS];
  __shared__ float aR2[10];
  __shared__ float bc[4];
  __shared__ int bi[4];
  int t = threadIdx.x;
  int numC = meta[0];
  if (t < 10){
    float l = anchors[3*t], w = anchors[3*t+1], h = anchors[3*t+2];
    aR2[t] = sqrtf(l*l + w*w + h*h) * 0.5f;
  }
  for (int p = t; p < NPTS; p += 1024){
    int id = order[p];
    sOrder[p] = id;
    oCx[p] = cx[id]; oCy[p] = cy[id]; oCz[p] = cz[id];
    oCls[p] = (short)cc[id];
    keep[p] = 1;
  }
  __syncthreads();
  for (int i = 0; i < numC; ++i){
    if (t == 0){
      int idx = sOrder[i];
      int cls = (int)oCls[i];
      bi[0] = idx;
      bi[1] = keep[i] ? 1 : 0;
      bi[2] = cls;
      bc[0] = oCx[i]; bc[1] = oCy[i]; bc[2] = oCz[i];
      bc[3] = (cls >= 0 && cls < 10) ? aR2[cls] : 0.0f;
    }
    __syncthreads();
    if (bi[1]){
      int idx = bi[0];
      short cls = (short)bi[2];
      float r2 = bc[3], mx = bc[0], my = bc[1], mz = bc[2];
      for (int p = i + 1 + t; p < numC; p += 1024){
        if (keep[p] && oCls[p] == cls){
          float ex = oCx[p]-mx, ey = oCy[p]-my, ez = oCz[p]-mz;
          if (sqrtf(ex*ex + ey*ey + ez*ez) < r2){
            keep[p] = 0;
            remap[sOrder[p]] = idx;   // absorbers are never absorbed later
          }
        }
      }
    }
    __syncthreads();
  }
}

__global__ void writeout(const int* __restrict__ dl, const int* __restrict__ remap,
                         float* __restrict__ out, int n){
  int j = blockIdx.x * blockDim.x + threadIdx.x;
  if (j < n) out[j] = (float)remap[dl[j]];
}

// ---------------------------------------------------------------------------
extern "C" void kernel_launch(void* const* d_in, const int* in_sizes, int n_in,
                              void* d_out, int out_size, void* d_ws, size_t ws_size,
                              hipStream_t stream){
  const float* pts  = (const float*)d_in[0];
  const int*   lbl  = (const int*)  d_in[1];
  const float* anch = (const float*)d_in[2];
  const int n = NPTS;   // in_sizes[0] == 3*3072

  char* p = (char*)d_ws;
  auto alloc = [&](size_t bytes){ void* r = (void*)p; p += (bytes + 255) & ~(size_t)255; return r; };
  int*   dl     = (int*)  alloc(sizeof(int)   * n);
  int*   ccA    = (int*)  alloc(sizeof(int)   * n);
  int*   counts = (int*)  alloc(sizeof(int)   * n);
  float* sx     = (float*)alloc(sizeof(float) * n);
  float* sy     = (float*)alloc(sizeof(float) * n);
  float* sz     = (float*)alloc(sizeof(float) * n);
  float* cxv    = (float*)alloc(sizeof(float) * n);
  float* cyv    = (float*)alloc(sizeof(float) * n);
  float* czv    = (float*)alloc(sizeof(float) * n);
  int*   order  = (int*)  alloc(sizeof(int)   * n);
  int*   remap  = (int*)  alloc(sizeof(int)   * n);
  int*   meta   = (int*)  alloc(sizeof(int)   * 16);
  float* n2     = (float*)alloc(sizeof(float) * n);
  float* nrm    = (float*)alloc(sizeof(float) * n);

  const size_t matOff = (size_t)1 << 20;           // matrices at +1MB
  const size_t matBytes = 2ull * n * n * sizeof(float);
  bool pre = (ws_size >= matOff + matBytes);
  float* dists  = (float*)((char*)d_ws + matOff);
  float* angles = dists + (size_t)n * n;

  int blocks = (n + 255) / 256;
  prep_norms<<<blocks, 256, 0, stream>>>(pts, n2, nrm, n);

  if (pre){
    dim3 grid(n / 16, n / (16 * 8));               // 192 x 24, 8 waves/block
    pair_wmma<<<grid, 256, 0, stream>>>(pts, n2, nrm, dists, angles, n);
    scan_kernel<true ><<<1, NTH, 0, stream>>>(pts, lbl, anch, dl, ccA, meta, dists, angles);
  } else {
    scan_kernel<false><<<1, NTH, 0, stream>>>(pts, lbl, anch, dl, ccA, meta, nullptr, nullptr);
  }

  zero_init  <<<blocks, 256, 0, stream>>>(counts, sx, sy, sz, remap, n);
  stats_accum<<<blocks, 256, 0, stream>>>(dl, pts, counts, sx, sy, sz, n);
  centres_fin<<<blocks, 256, 0, stream>>>(counts, sx, sy, sz, cxv, cyv, czv, n);
  sort_kernel<<<1, NTH, 0, stream>>>(counts, meta, order, n);
  merge_kernel<<<1, NTH, 0, stream>>>(order, cxv, cyv, czv, ccA, anch, meta, remap);
  writeout   <<<blocks, 256, 0, stream>>>(dl, remap, (float*)d_out, n);
}